// RealEntropyCodec_23398981829012
// MI455X (gfx1250) — compile-verified
//
#include <hip/hip_runtime.h>
#include <hip/hip_bf16.h>

typedef float v2f __attribute__((ext_vector_type(2)));
typedef float v8f __attribute__((ext_vector_type(8)));

#define HIST_WGS 64          // workgroups building private LDS histograms (rows of P)
#define THREADS  1024        // hist + finalize block size
#define RTHREADS 256         // wmma-reduce block size (low VGPR pressure)

// ---------------------------------------------------------------------------
// Kernel 1: per-workgroup private histogram in LDS (full 64K bins = 256 KB,
// only possible on CDNA5's 320 KB WGP LDS), flushed as one float row of the
// partials matrix P[HIST_WGS][K]. No global atomics anywhere.
// ---------------------------------------------------------------------------
__global__ __launch_bounds__(THREADS) void hist_lds_kernel(
    const int* __restrict__ idx, int n, int K, float* __restrict__ P) {
  extern __shared__ unsigned int lh[];   // K u32 bins (dynamic LDS, K*4 bytes)
  const int tid = threadIdx.x;

  // zero private histogram
  for (int i = tid; i < K; i += THREADS) lh[i] = 0u;
  __syncthreads();

  // stream indices: grid-stride over int4 vectors, ds_add_u32 per element
  const int nvec   = n >> 2;
  const int stride = gridDim.x * THREADS;
  const int4* in4  = (const int4*)idx;
  for (int i = blockIdx.x * THREADS + tid; i < nvec; i += stride) {
    __builtin_prefetch(&in4[i + stride], 0, 0);   // global_prefetch_b8 lookahead
    int4 v = in4[i];
    atomicAdd(&lh[(unsigned)v.x], 1u);
    atomicAdd(&lh[(unsigned)v.y], 1u);
    atomicAdd(&lh[(unsigned)v.z], 1u);
    atomicAdd(&lh[(unsigned)v.w], 1u);
  }
  // tail (n not multiple of 4): handled by block 0 only
  if (blockIdx.x == 0 && tid < (n & 3)) {
    atomicAdd(&lh[(unsigned)idx[(n & ~3) + tid]], 1u);
  }
  __syncthreads();

  // flush private histogram as one float row of P (vectorized b128 stores)
  float* row = P + (size_t)blockIdx.x * (size_t)K;
  for (int i = tid * 4; i < K; i += THREADS * 4) {
    uint4 v = *(const uint4*)&lh[i];
    float4 f;
    f.x = (float)v.x; f.y = (float)v.y; f.z = (float)v.z; f.w = (float)v.w;
    *(float4*)&row[i] = f;
  }
}

// ---------------------------------------------------------------------------
// Kernel 2: reduce the 64 partial histograms per bin on the MATRIX pipe:
//   hist = ones(1x64) @ P(64xK)  via chained V_WMMA_F32_16X16X4_F32.
// A = 16x4 of ones; B = 4x16 tile of P (K-rows striped 2-per-half-wave across
// 2 VGPRs per the CDNA5 B-matrix layout). All 16 B-tiles are preloaded as one
// batch of 32 global loads (single loadcnt wait), then 16 WMMAs chain C->D.
// Every row of D equals the 16 column sums; lanes 0..15, VGPR0 hold them.
// ---------------------------------------------------------------------------
__global__ __launch_bounds__(RTHREADS) void reduce_partials_wmma(
    const float* __restrict__ P, float* __restrict__ hist, int K) {
  const int gt     = blockIdx.x * RTHREADS + threadIdx.x;
  const int wave   = gt >> 5;
  const int lane   = gt & 31;
  const int nWaves = (gridDim.x * RTHREADS) >> 5;
  const int nTiles = K >> 4;                  // 16 bins per tile

  v2f a; a.x = 1.0f; a.y = 1.0f;              // all-ones A (16x4)
  const size_t rowOff = (size_t)((lane >> 4) << 1) * (size_t)K;  // +0 or +2K

  for (int tile = wave; tile < nTiles; tile += nWaves) {
    const int col = (tile << 4) + (lane & 15);
    // per-lane base: row ((lane>=16)?2:0), column col
    const float* base = P + rowOff + (size_t)col;

    // batch-load all 16 B-tiles (32 outstanding global_load_b32)
    v2f b[HIST_WGS >> 2];
    #pragma unroll
    for (int step = 0; step < (HIST_WGS >> 2); ++step) {
      const float* p = base + (size_t)(step << 2) * (size_t)K;
      b[step].x = p[0];
      b[step].y = p[(size_t)K];
    }

    // chained matrix-pipe accumulation
    v8f c = {0.f, 0.f, 0.f, 0.f, 0.f, 0.f, 0.f, 0.f};
    #pragma unroll
    for (int step = 0; step < (HIST_WGS >> 2); ++step) {
      c = __builtin_amdgcn_wmma_f32_16x16x4_f32(
          /*neg_a=*/false, a, /*neg_b=*/false, b[step],
          /*c_mod=*/(short)0, c, /*reuse_a=*/false, /*reuse_b=*/false);
    }
    if (lane < 16) hist[col] = c[0];          // D row 0 = column sums
  }
}

// ---------------------------------------------------------------------------
// Kernel 3: finalize. Single 1024-thread block, two fixed-order passes over
// L2-hot hist + symbol_counts:
//   total = sum((sc[k]+hist[k]) + 1e-8)
//   out   = (1/N) * sum( hist[k] * -log2(max(((sc[k]+hist[k])+1e-8)/total, 1e-10)) )
// ---------------------------------------------------------------------------
__global__ __launch_bounds__(THREADS) void finalize_kernel(
    const float* __restrict__ hist, const float* __restrict__ sc,
    int K, int n, float* __restrict__ out) {
  __shared__ float red[THREADS];
  const int tid = threadIdx.x;

  float s = 0.f;
  for (int k = tid; k < K; k += THREADS) s += (sc[k] + hist[k]) + 1e-8f;
  red[tid] = s;
  __syncthreads();
  for (int off = THREADS >> 1; off > 0; off >>= 1) {
    if (tid < off) red[tid] += red[tid + off];
    __syncthreads();
  }
  const float inv_total = 1.0f / red[0];
  __syncthreads();

  float acc = 0.f;
  for (int k = tid; k < K; k += THREADS) {
    const float h  = hist[k];
    const float sm = (sc[k] + h) + 1e-8f;
    float p = sm * inv_total;
    p = fmaxf(p, 1e-10f);
    acc += h * (-__log2f(p));
  }
  red[tid] = acc;
  __syncthreads();
  for (int off = THREADS >> 1; off > 0; off >>= 1) {
    if (tid < off) red[tid] += red[tid + off];
    __syncthreads();
  }
  if (tid == 0) out[0] = red[0] / (float)n;
}

// ---------------------------------------------------------------------------
extern "C" void kernel_launch(void* const* d_in, const int* in_sizes, int n_in,
                              void* d_out, int out_size, void* d_ws, size_t ws_size,
                              hipStream_t stream) {
  const int*   idx = (const int*)d_in[0];     // (B*T,) int32 indices
  const float* sc  = (const float*)d_in[1];   // (K,)   float32 symbol_counts
  const int n = in_sizes[0];                  // 16,777,216
  const int K = in_sizes[1];                  // 65,536
  float* out  = (float*)d_out;                // scalar

  // workspace layout: [ hist: K floats ][ P: HIST_WGS * K floats ]
  float* hist = (float*)d_ws;
  float* P    = hist + (size_t)K;

  // 1) 64 WGPs, each with a full 256 KB private LDS histogram
  hist_lds_kernel<<<HIST_WGS, THREADS, (size_t)K * sizeof(unsigned int), stream>>>(
      idx, n, K, P);

  // 2) matrix-pipe reduction of the 64xK partials matrix (ones @ P)
  reduce_partials_wmma<<<256, RTHREADS, 0, stream>>>(P, hist, K);

  // 3) deterministic scalar finalize
  finalize_kernel<<<1, THREADS, 0, stream>>>(hist, sc, K, n, out);
}